// Model_66340064854704
// MI455X (gfx1250) — compile-verified
//
#include <hip/hip_runtime.h>
#include <math.h>

// ---------------------------------------------------------------------------
// Model dims
#define B_   32
#define T_   12
#define N_   512
#define H_   64
#define MT_  60
#define HOR_ 3
#define LRA  0.2f

typedef float v2f __attribute__((ext_vector_type(2)));
typedef float v8f __attribute__((ext_vector_type(8)));

#define ACT_NONE     0
#define ACT_SIG      1
#define ACT_TANH     2
#define ACT_RELU     3
#define ACT_LRELU02  4
#define ACT_LRELU001 5
#define ACT_EXPNRELU 6

__device__ __forceinline__ float actf(float x, int act) {
  switch (act) {
    case ACT_SIG:      return 1.f / (1.f + __expf(-x));
    case ACT_TANH:     return tanhf(x);
    case ACT_RELU:     return fmaxf(x, 0.f);
    case ACT_LRELU02:  return x > 0.f ? x : LRA  * x;
    case ACT_LRELU001: return x > 0.f ? x : 0.01f * x;
    case ACT_EXPNRELU: return __expf(-fmaxf(x, 0.f));
    default:           return x;
  }
}

__device__ __forceinline__ v8f vzero8() {
  v8f z = {0.f,0.f,0.f,0.f,0.f,0.f,0.f,0.f};
  return z;
}

#define WMMA_F32X4(a, b, c) \
  __builtin_amdgcn_wmma_f32_16x16x4_f32(false, (a), false, (b), (short)0, (c), false, false)

// ---------------------------------------------------------------------------
// WMMA f32 16x16x4 tile kernel (layouts per CDNA5 ISA 7.12.2).
// Preconditions exploited (all true for this model): M % 16 == 0, K % 4 == 0,
// lda even (8B-aligned v2f loads). N edges handled by clamping the column
// (branch-free; epilogue masks the store), so the K loop has no guards at all:
//   1x b64 A-load + 2x b32 B-loads + v_wmma per step.
__device__ __forceinline__ v8f gemm_tile(v8f acc, const float* __restrict__ A,
                                         const float* __restrict__ Bp,
                                         int lda, int ldb, int N, int K,
                                         int m0, int n0) {
  const int lane = threadIdx.x & 31;
  const int half2 = (lane >> 4) << 1;          // 0 or 2
  const int r = m0 + (lane & 15);
  int c = n0 + (lane & 15);
  if (c > N - 1) c = N - 1;                    // clamp: loads stay in-bounds
  const float* ap = A + (long)r * lda + half2; // A[r][k0+half2 ..]
  const float* bp = Bp + (long)half2 * ldb + c;
  const long bstep = 4L * ldb;
  __builtin_prefetch(bp, 0, 1);                // global_prefetch_b8 on B panel
  for (int k0 = 0; k0 < K; k0 += 4) {
    v2f a = *(const v2f*)ap;                   // K{0,1} / K{2,3}
    v2f b;
    b.x = bp[0];
    b.y = bp[ldb];
    acc = WMMA_F32X4(a, b, acc);
    ap += 4;
    bp += bstep;
  }
  return acc;
}

// C/D 16x16: VGPR v: lanes<16 -> row m0+v ; lanes>=16 -> row m0+v+8
__device__ __forceinline__ void stC(float* C, int ldc, int N, int m0, int n0,
                                    v8f acc, const float* bias,
                                    float alpha, float beta, int act) {
  const int lane = threadIdx.x & 31;
  const int c  = n0 + (lane & 15);
  const int rb = m0 + ((lane >> 4) << 3);
  if (c < N) {
    float bv = bias ? bias[c] : 0.f;
    float* cp = C + (long)rb * ldc + c;
#pragma unroll
    for (int v = 0; v < 8; ++v) {
      float x = alpha * acc[v] + bv;
      if (beta != 0.f) x += beta * cp[0];
      cp[0] = actf(x, act);
      cp += ldc;
    }
  }
}

// ---------------------------------------------------------------------------
// Generic batched GEMM: C = act(alpha*A@B + bias + beta*C); 8 waves = 8 tiles
__global__ __launch_bounds__(256) void k_gemm(
    const float* __restrict__ A, const float* __restrict__ Bp,
    const float* __restrict__ bias, float* C,
    int M, int N, int K, int lda, int ldb, int ldc,
    long sA, long sB, long sC, float alpha, float beta, int act) {
  long z = blockIdx.z;
  A += z * sA;  Bp += z * sB;  C += z * sC;
  const int tilesN = (N + 15) >> 4;
  const int tilesM = M >> 4;                 // M % 16 == 0 by construction
  const int tile = blockIdx.x * 8 + (threadIdx.x >> 5);
  if (tile >= tilesM * tilesN) return;       // wave-uniform exit
  const int tm = tile / tilesN, tn = tile % tilesN;
  v8f acc = gemm_tile(vzero8(), A, Bp, lda, ldb, N, K, tm * 16, tn * 16);
  stC(C, ldc, N, tm * 16, tn * 16, acc, bias, alpha, beta, act);
}

// ---------------------------------------------------------------------------
// GRU-D scan: one persistent workgroup, state in LDS, WMMA per-step GEMMs
__global__ __launch_bounds__(1024) void k_grud(
    const float* __restrict__ x, const float* __restrict__ xmean,
    const float* __restrict__ lastx,
    const float* gxw, const float* gxb, const float* ghw, const float* ghb,
    const float* zwx, const float* zwh, const float* zwm, const float* zb,
    const float* rwx, const float* rwh, const float* rwm, const float* rb,
    const float* hwx, const float* hwh, const float* hwm, const float* hb,
    const float* outw, const float* outb, float* gd) {
  extern __shared__ float sm[];
  float* s_xlast = sm;                    // 32*512
  float* s_xhat  = s_xlast + B_ * N_;     // 32*512
  float* s_gx    = s_xhat  + B_ * N_;     // 32*512
  float* s_stage = s_gx    + B_ * N_;     // 32*512 (delta then mask)
  float* s_h     = s_stage + B_ * N_;     // 32*64
  float* s_gh    = s_h  + B_ * H_;        // 32*64 (reused as r*h)
  float* s_z     = s_gh + B_ * H_;
  float* s_r     = s_z  + B_ * H_;
  float* s_ht    = s_r  + B_ * H_;

  const int tid = threadIdx.x;
  const int wid = tid >> 5;

  for (int i = tid; i < B_ * N_; i += 1024) s_xlast[i] = lastx[i];
  for (int i = tid; i < B_ * H_; i += 1024) s_h[i] = 0.f;
  __syncthreads();

  for (int t = 0; t < T_; ++t) {
    // stage delta
    for (int i = tid; i < B_ * N_; i += 1024) {
      int b = i >> 9, n = i & (N_ - 1);
      s_stage[i] = x[(((long)b * T_ + t) * N_ + n) * 3 + 2];
    }
    __syncthreads();
    // g_x = exp(-relu(delta@Wgx+b)) : 2x32 tiles; g_h : 2x4 tiles
    for (int j = wid; j < 64; j += 32) {
      int tm = j >> 5, tn = j & 31;
      v8f acc = gemm_tile(vzero8(), s_stage, gxw, N_, N_, N_, N_, tm*16, tn*16);
      stC(s_gx, N_, N_, tm*16, tn*16, acc, gxb, 1.f, 0.f, ACT_EXPNRELU);
    }
    if (wid < 8) {
      int tm = wid >> 2, tn = wid & 3;
      v8f acc = gemm_tile(vzero8(), s_stage, ghw, N_, H_, H_, N_, tm*16, tn*16);
      stC(s_gh, H_, H_, tm*16, tn*16, acc, ghb, 1.f, 0.f, ACT_EXPNRELU);
    }
    __syncthreads();
    // x_hat, x_last update, stage mask; h *= g_h
    for (int i = tid; i < B_ * N_; i += 1024) {
      int b = i >> 9, n = i & (N_ - 1);
      long base = (((long)b * T_ + t) * N_ + n) * 3;
      float xv = x[base], m = x[base + 1];
      float gx = s_gx[i], xl = s_xlast[i];
      s_xhat[i]  = m * xv + (1.f - m) * (gx * xl + (1.f - gx) * xmean[n]);
      s_xlast[i] = m * xv + (1.f - m) * xl;
      s_stage[i] = m;
    }
    for (int i = tid; i < B_ * H_; i += 1024) s_h[i] *= s_gh[i];
    __syncthreads();
    // z (waves 0-7), r (waves 8-15): K = 512 + 64 + 512
    if (wid < 16) {
      int g = wid >> 3, j = wid & 7;
      int tm = j >> 2, tn = j & 3;
      const float* Wx = g ? rwx : zwx;
      const float* Wh = g ? rwh : zwh;
      const float* Wm = g ? rwm : zwm;
      const float* bb = g ? rb  : zb;
      v8f acc = vzero8();
      acc = gemm_tile(acc, s_xhat,  Wx, N_, H_, H_, N_, tm*16, tn*16);
      acc = gemm_tile(acc, s_h,     Wh, H_, H_, H_, H_, tm*16, tn*16);
      acc = gemm_tile(acc, s_stage, Wm, N_, H_, H_, N_, tm*16, tn*16);
      stC(g ? s_r : s_z, H_, H_, tm*16, tn*16, acc, bb, 1.f, 0.f, ACT_SIG);
    }
    __syncthreads();
    for (int i = tid; i < B_ * H_; i += 1024) s_gh[i] = s_r[i] * s_h[i];
    __syncthreads();
    if (wid < 8) {
      int tm = wid >> 2, tn = wid & 3;
      v8f acc = vzero8();
      acc = gemm_tile(acc, s_xhat,  hwx, N_, H_, H_, N_, tm*16, tn*16);
      acc = gemm_tile(acc, s_gh,    hwh, H_, H_, H_, H_, tm*16, tn*16);
      acc = gemm_tile(acc, s_stage, hwm, N_, H_, H_, N_, tm*16, tn*16);
      stC(s_ht, H_, H_, tm*16, tn*16, acc, hb, 1.f, 0.f, ACT_TANH);
    }
    __syncthreads();
    for (int i = tid; i < B_ * H_; i += 1024) {
      float z = s_z[i];
      s_h[i] = (1.f - z) * s_h[i] + z * s_ht[i];
    }
    __syncthreads();
    // out_t = h @ Wout + bout -> gd[t] (layout t,b,n)
    for (int j = wid; j < 64; j += 32) {
      int tm = j >> 5, tn = j & 31;
      v8f acc = gemm_tile(vzero8(), s_h, outw, H_, N_, N_, H_, tm*16, tn*16);
      stC(gd + (long)t * B_ * N_, N_, N_, tm*16, tn*16, acc, outb, 1.f, 0.f, ACT_NONE);
    }
    __syncthreads();
  }
}

// ---------------------------------------------------------------------------
// GRU scan over 512 nodes; h in LDS; fused key/query accumulation
__global__ __launch_bounds__(1024) void k_gru(
    const float* __restrict__ gd, const float* __restrict__ wi,
    const float* __restrict__ wh, const float* __restrict__ bi,
    const float* __restrict__ bh,
    const float* __restrict__ wk, const float* __restrict__ wq,
    float* gi, float* gh, float* keyo, float* qryo) {
  extern __shared__ float sm[];
  float* s_h   = sm;                  // 32*512
  float* s_key = s_h   + B_ * N_;     // 32*512
  float* s_qry = s_key + B_ * N_;     // 32*512
  float* s_xt  = s_qry + B_ * N_;     // 32*12

  const int tid = threadIdx.x;
  const int wid = tid >> 5;
  for (int i = tid; i < B_ * N_; i += 1024) { s_h[i] = 0.f; s_key[i] = 0.f; s_qry[i] = 0.f; }
  __syncthreads();

  for (int n = 0; n < N_; ++n) {
    for (int i = tid; i < B_ * T_; i += 1024) {
      int b = i / T_, t = i % T_;
      s_xt[i] = gd[((long)t * B_ + b) * N_ + n];
    }
    __syncthreads();
    // gi = xt @ wi + bi (K=12, VALU)
    for (int i = tid; i < B_ * 3 * N_; i += 1024) {
      int b = i / (3 * N_), j = i % (3 * N_);
      float s = bi[j];
#pragma unroll
      for (int t = 0; t < T_; ++t) s += s_xt[b * T_ + t] * wi[t * 3 * N_ + j];
      gi[i] = s;
    }
    // gh = h @ wh + bh : 2x96 tiles = 192, K=512, via WMMA
    for (int j = wid; j < 192; j += 32) {
      int tm = j / 96, tn = j % 96;
      v8f acc = gemm_tile(vzero8(), s_h, wh, N_, 3 * N_, 3 * N_, N_, tm*16, tn*16);
      stC(gh, 3 * N_, 3 * N_, tm*16, tn*16, acc, bh, 1.f, 0.f, ACT_NONE);
    }
    __threadfence();
    __syncthreads();
    for (int i = tid; i < B_ * N_; i += 1024) {
      int b = i >> 9, k = i & (N_ - 1);
      long o = (long)b * 3 * N_ + k;
      float r  = 1.f / (1.f + __expf(-(gi[o] + gh[o])));
      float z  = 1.f / (1.f + __expf(-(gi[o + N_] + gh[o + N_])));
      float nn = tanhf(gi[o + 2 * N_] + r * gh[o + 2 * N_]);
      float hv = (1.f - z) * nn + z * s_h[i];
      s_h[i] = hv;
      s_key[i] += hv * wk[n];
      s_qry[i] += hv * wq[n];
    }
    __syncthreads();
  }
  for (int i = tid; i < B_ * N_; i += 1024) { keyo[i] = s_key[i]; qryo[i] = s_qry[i]; }
}

// ---------------------------------------------------------------------------
// att[i][j] = mean_b softmax_j( leaky_relu(key[b,i]+query[b,j], 0.2) )
__global__ void k_att(const float* __restrict__ key, const float* __restrict__ qry,
                      float* att) {
  __shared__ float red[256];
  const int i = blockIdx.x, tid = threadIdx.x;
  float acc0 = 0.f, acc1 = 0.f;
  for (int b = 0; b < B_; ++b) {
    float kb = key[b * N_ + i];
    float d0 = kb + qry[b * N_ + tid];
    float d1 = kb + qry[b * N_ + tid + 256];
    d0 = d0 > 0.f ? d0 : LRA * d0;
    d1 = d1 > 0.f ? d1 : LRA * d1;
    red[tid] = fmaxf(d0, d1);
    __syncthreads();
    for (int s = 128; s > 0; s >>= 1) {
      if (tid < s) red[tid] = fmaxf(red[tid], red[tid + s]);
      __syncthreads();
    }
    float mx = red[0];
    __syncthreads();
    float e0 = __expf(d0 - mx), e1 = __expf(d1 - mx);
    red[tid] = e0 + e1;
    __syncthreads();
    for (int s = 128; s > 0; s >>= 1) {
      if (tid < s) red[tid] += red[tid + s];
      __syncthreads();
    }
    float inv = 1.f / red[0];
    __syncthreads();
    acc0 += e0 * inv * (1.f / B_);
    acc1 += e1 * inv * (1.f / B_);
  }
  att[(long)i * N_ + tid]       = acc0;
  att[(long)i * N_ + tid + 256] = acc1;
}

__global__ void k_degree(const float* att, float* deg, float* dhat) {
  __shared__ float red[256];
  const int i = blockIdx.x, tid = threadIdx.x;
  float s = 0.f;
  for (int j = tid; j < N_; j += 256) s += att[(long)i * N_ + j];
  red[tid] = s;
  __syncthreads();
  for (int st = 128; st > 0; st >>= 1) {
    if (tid < st) red[tid] += red[tid + st];
    __syncthreads();
  }
  if (tid == 0) { deg[i] = red[0]; dhat[i] = 1.f / (sqrtf(red[0]) + 1e-7f); }
}

__global__ void k_lap(const float* att, const float* deg, const float* dhat,
                      float* lap) {
  long i = (long)blockIdx.x * blockDim.x + threadIdx.x;
  if (i >= (long)N_ * N_) return;
  int r = (int)(i >> 9), c = (int)(i & (N_ - 1));
  float a = 0.5f * (att[i] + att[(long)c * N_ + r]);
  float v = ((r == c) ? deg[r] : 0.f) - a;
  lap[i] = dhat[r] * v * dhat[c];
}

// ---------------------------------------------------------------------------
__global__ void k_fill(float* p, float v, long n) {
  long i = (long)blockIdx.x * blockDim.x + threadIdx.x;
  if (i < n) p[i] = v;
}
__global__ void k_mul(float* a, const float* b, long n) {
  long i = (long)blockIdx.x * blockDim.x + threadIdx.x;
  if (i < n) a[i] *= b[i];
}
__global__ void k_extractX(const float* x, float* X) {
  long i = (long)blockIdx.x * blockDim.x + threadIdx.x;
  if (i >= (long)B_ * N_ * T_) return;
  int t = (int)(i % T_);
  long r = i / T_;
  int n = (int)(r & (N_ - 1)), b = (int)(r >> 9);
  X[i] = x[(((long)b * T_ + t) * N_ + n) * 3 + 0];
}
// 4-point FFT over k (a0 plane is zero by construction of mul_L[0])
__global__ void k_fft4(const float* gfted, float* re, float* im) {
  long i = (long)blockIdx.x * blockDim.x + threadIdx.x;
  if (i >= (long)B_ * N_ * T_) return;
  int t = (int)(i % T_);
  long r = i / T_;
  long s = (long)B_ * N_ * T_;
  float a0 = gfted[i], a1 = gfted[i + s], a2 = gfted[i + 2 * s], a3 = gfted[i + 3 * s];
  long o = r * 48 + t;
  re[o]      = a0 + a1 + a2 + a3;
  re[o + 12] = a0 - a2;
  re[o + 24] = a0 - a1 + a2 - a3;
  re[o + 36] = a0 - a2;
  im[o]      = 0.f;
  im[o + 12] = a3 - a1;
  im[o + 24] = 0.f;
  im[o + 36] = a1 - a3;
}
// irfft(n=4) closed form; irfft ignores imag of DC & Nyquist
__global__ void k_irfft4(const float* re, const float* im, float* gci) {
  long i = (long)blockIdx.x * blockDim.x + threadIdx.x;
  if (i >= (long)B_ * N_ * MT_) return;
  int m = (int)(i % MT_);
  long r = i / MT_;
  long base = r * 240 + m;
  float re0 = re[base], re1 = re[base + 60], re2 = re[base + 120];
  float im1 = im[base + 60];
  float* o = gci + base;
  o[0]   = 0.25f * (re0 + 2.f * re1 + re2);
  o[60]  = 0.25f * (re0 - 2.f * im1 - re2);
  o[120] = 0.25f * (re0 - 2.f * re1 + re2);
  o[180] = 0.25f * (re0 + 2.f * im1 - re2);
}
// out[b,n,o] = act( sum_{k=1..3} sum_f y[k,b,n,f] * w[k,f,o] )   (k=0 plane is zero)
__global__ void k_gcomb(const float* y, const float* w, float* out,
                        int F, int O, int act) {
  long i = (long)blockIdx.x * blockDim.x + threadIdx.x;
  if (i >= (long)B_ * N_ * O) return;
  int o = (int)(i % O);
  long r = i / O;
  float s = 0.f;
  for (int k = 1; k < 4; ++k)
    for (int f = 0; f < F; ++f)
      s += y[((long)k * B_ * N_ + r) * F + f] * w[(k * F + f) * O + o];
  out[i] = actf(s, act);
}
// (b,n,h) -> (b,h,n)
__global__ void k_outT(const float* src, float* dst) {
  long i = (long)blockIdx.x * blockDim.x + threadIdx.x;
  if (i >= (long)B_ * N_ * HOR_) return;
  int h = (int)(i % HOR_);
  long r = i / HOR_;
  int n = (int)(r & (N_ - 1)), b = (int)(r >> 9);
  dst[((long)b * HOR_ + h) * N_ + n] = src[i];
}

// ---------------------------------------------------------------------------
static inline void gemm(hipStream_t s, const float* A, const float* Bp,
                        const float* bias, float* C, int M, int N, int K,
                        int lda, int ldb, int ldc, long sA, long sB, long sC,
                        int batch, float alpha, float beta, int act) {
  int tiles = (M >> 4) * ((N + 15) >> 4);
  dim3 g((unsigned)((tiles + 7) / 8), 1, (unsigned)batch);
  k_gemm<<<g, 256, 0, s>>>(A, Bp, bias, C, M, N, K, lda, ldb, ldc,
                           sA, sB, sC, alpha, beta, act);
}
static inline dim3 g1(long n) { return dim3((unsigned)((n + 255) / 256)); }

extern "C" void kernel_launch(void* const* d_in, const int* in_sizes, int n_in,
                              void* d_out, int out_size, void* d_ws, size_t ws_size,
                              hipStream_t stream) {
  (void)in_sizes; (void)n_in; (void)ws_size; (void)out_size;
  auto F = [&](int i) { return (const float*)d_in[i]; };
  const float* x = F(0); const float* xmean = F(1); const float* lastx = F(2);
  const float* wkey = F(3); const float* wquery = F(4);
  const float* gxw = F(5), *gxb = F(6), *ghw = F(7), *ghb = F(8);
  const float* zwx = F(9), *zwh = F(10), *zwm = F(11), *zb = F(12);
  const float* rwx = F(13), *rwh = F(14), *rwm = F(15), *rb = F(16);
  const float* hwx = F(17), *hwh = F(18), *hwm = F(19), *hb = F(20);
  const float* outw = F(21), *outb = F(22);
  const float* gwi = F(23), *gwh = F(24), *gbi = F(25), *gbh = F(26);
  struct Blk { const float *wt, *fw, *fb, *frw, *frb, *bsw, *bsb, *gl[6][4], *bw, *bb; };
  Blk blk[2]; int q = 27;
  for (int si = 0; si < 2; ++si) {
    blk[si].wt = F(q++); blk[si].fw = F(q++); blk[si].fb = F(q++);
    blk[si].frw = F(q++); blk[si].frb = F(q++);
    blk[si].bsw = F(q++); blk[si].bsb = F(q++);
    for (int g = 0; g < 6; ++g) for (int e = 0; e < 4; ++e) blk[si].gl[g][e] = F(q++);
    blk[si].bw = nullptr; blk[si].bb = nullptr;
    if (si == 0) { blk[si].bw = F(q++); blk[si].bb = F(q++); }
  }
  const float* fc1w = F(q++), *fc1b = F(q++), *fc2w = F(q++), *fc2b = F(q++);
  const float* gc1 = F(q++), *gc2 = F(q++);

  // workspace partition (floats)
  float* w = (float*)d_ws; long off = 0;
  auto alloc = [&](long n) { float* r = w + off; off += n; return r; };
  float* gd    = alloc((long)T_ * B_ * N_);
  float* keyb  = alloc(B_ * N_);
  float* qryb  = alloc(B_ * N_);
  float* attb  = alloc((long)N_ * N_);
  float* degb  = alloc(N_);
  float* dhatb = alloc(N_);
  float* lapb  = alloc((long)N_ * N_);
  float* mulL  = alloc(4L * N_ * N_);
  float* gib   = alloc(B_ * 3 * N_);
  float* ghb2  = alloc(B_ * 3 * N_);
  float* Xbuf  = alloc((long)B_ * N_ * T_);
  float* gfted = alloc(4L * B_ * N_ * T_);
  float* re48  = alloc((long)B_ * N_ * 48);
  float* im48  = alloc((long)B_ * N_ * 48);
  float* Bu[4];
  for (int i = 0; i < 4; ++i) Bu[i] = alloc((long)B_ * N_ * 240);
  float* gci    = alloc((long)B_ * N_ * 240);
  float* igf    = alloc((long)B_ * N_ * MT_);
  float* fsrc   = alloc((long)B_ * N_ * MT_);
  float* fcsum  = alloc((long)B_ * N_ * T_);
  float* shortb = alloc((long)B_ * N_ * T_);
  float* fctmp  = alloc((long)B_ * N_ * T_);
  float* Ff     = alloc((long)B_ * N_ * HOR_);
  float* ybuf   = alloc(4L * B_ * N_ * 16);
  float* gout1  = alloc((long)B_ * N_ * 16);
  float* gout2  = alloc((long)B_ * N_ * HOR_);

  const long NN = (long)N_ * N_;

  // 1) GRU-D scan (persistent, LDS-resident state, WMMA)
  size_t smem_grud = (size_t)(4 * B_ * N_ + 5 * B_ * H_) * sizeof(float);
  k_grud<<<1, 1024, smem_grud, stream>>>(x, xmean, lastx,
      gxw, gxb, ghw, ghb, zwx, zwh, zwm, zb, rwx, rwh, rwm, rb,
      hwx, hwh, hwm, hb, outw, outb, gd);

  // 2) GRU scan + fused key/query accumulation
  size_t smem_gru = (size_t)(3 * B_ * N_ + B_ * T_) * sizeof(float);
  k_gru<<<1, 1024, smem_gru, stream>>>(gd, gwi, gwh, gbi, gbh, wkey, wquery,
                                       gib, ghb2, keyb, qryb);

  // 3) attention -> Laplacian -> Chebyshev stack mul_L
  k_att<<<N_, 256, 0, stream>>>(keyb, qryb, attb);
  k_degree<<<N_, 256, 0, stream>>>(attb, degb, dhatb);
  k_lap<<<g1(NN), 256, 0, stream>>>(attb, degb, dhatb, lapb);
  k_fill<<<g1(NN), 256, 0, stream>>>(mulL, 0.f, NN);
  hipMemcpyAsync(mulL + NN, lapb, NN * sizeof(float), hipMemcpyDeviceToDevice, stream);
  gemm(stream, lapb, lapb, nullptr, mulL + 2 * NN, N_, N_, N_, N_, N_, N_,
       0, 0, 0, 1, 2.f, 0.f, ACT_NONE);                  // t3 = 2*lap@lap
  hipMemcpyAsync(mulL + 3 * NN, lapb, NN * sizeof(float), hipMemcpyDeviceToDevice, stream);
  gemm(stream, lapb, mulL + 2 * NN, nullptr, mulL + 3 * NN, N_, N_, N_, N_, N_, N_,
       0, 0, 0, 1, 2.f, -1.f, ACT_NONE);                 // 2*lap@t3 - lap

  // 4) stock blocks
  k_extractX<<<g1((long)B_ * N_ * T_), 256, 0, stream>>>(x, Xbuf);
  const float* Xcur = Xbuf;
  const long MNT = (long)B_ * N_ * T_;
  for (int si = 0; si < 2; ++si) {
    Blk& bk = blk[si];
    k_fill<<<g1(MNT), 256, 0, stream>>>(gfted, 0.f, MNT);   // k=0 plane
    for (int k = 1; k < 4; ++k)
      gemm(stream, mulL + (long)k * NN, Xcur, nullptr, gfted + (long)k * MNT,
           N_, T_, N_, N_, T_, T_, 0, (long)N_ * T_, (long)N_ * T_, B_,
           1.f, 0.f, ACT_NONE);
    k_fft4<<<g1(MNT), 256, 0, stream>>>(gfted, re48, im48);
    auto glu = [&](const float* in, int Kin, const float* const* gp,
                   float* L, float* R) {
      gemm(stream, in, gp[0], gp[1], L, B_ * N_, 240, Kin, Kin, 240, 240,
           0, 0, 0, 1, 1.f, 0.f, ACT_NONE);
      gemm(stream, in, gp[2], gp[3], R, B_ * N_, 240, Kin, Kin, 240, 240,
           0, 0, 0, 1, 1.f, 0.f, ACT_SIG);
      k_mul<<<g1((long)B_ * N_ * 240), 256, 0, stream>>>(L, R, (long)B_ * N_ * 240);
    };
    glu(re48, 48,  bk.gl[0], Bu[0], Bu[1]);
    glu(Bu[0], 240, bk.gl[2], Bu[1], Bu[2]);
    glu(Bu[1], 240, bk.gl[4], Bu[2], Bu[0]);   // real -> Bu[2]
    glu(im48, 48,  bk.gl[1], Bu[0], Bu[1]);
    glu(Bu[0], 240, bk.gl[3], Bu[1], Bu[3]);
    glu(Bu[1], 240, bk.gl[5], Bu[3], Bu[0]);   // imag -> Bu[3]
    k_irfft4<<<g1((long)B_ * N_ * MT_), 256, 0, stream>>>(Bu[2], Bu[3], gci);
    gemm(stream, gci, bk.wt, nullptr, igf, B_ * N_, MT_, 240, 240, MT_, MT_,
         0, 0, 0, 1, 1.f, 0.f, ACT_NONE);
    gemm(stream, igf, bk.fw, bk.fb, fsrc, B_ * N_, MT_, MT_, MT_, MT_, MT_,
         0, 0, 0, 1, 1.f, 0.f, ACT_SIG);
    gemm(stream, fsrc, bk.frw, bk.frb, fcsum, B_ * N_, T_, MT_, MT_, T_, T_,
         0, 0, 0, 1, 1.f, si == 0 ? 0.f : 1.f, ACT_NONE);
    if (si == 0) {
      gemm(stream, Xcur, bk.bsw, bk.bsb, shortb, B_ * N_, T_, T_, T_, T_, T_,
           0, 0, 0, 1, 1.f, 0.f, ACT_NONE);
      gemm(stream, igf, bk.bw, bk.bb, shortb, B_ * N_, T_, MT_, MT_, T_, T_,
           0, 0, 0, 1, 1.f, -1.f, ACT_SIG);   // sigmoid(backcast - short)
      Xcur = shortb;
    }
  }

  // 5) fc1/fc2 + graph convs + outputs
  gemm(stream, fcsum, fc1w, fc1b, fctmp, B_ * N_, T_, T_, T_, T_, T_,
       0, 0, 0, 1, 1.f, 0.f, ACT_LRELU001);
  gemm(stream, fctmp, fc2w, fc2b, Ff, B_ * N_, HOR_, T_, T_, HOR_, HOR_,
       0, 0, 0, 1, 1.f, 0.f, ACT_NONE);
  for (int k = 1; k < 4; ++k)
    gemm(stream, mulL + (long)k * NN, Ff, nullptr, ybuf + (long)k * B_ * N_ * HOR_,
         N_, HOR_, N_, N_, HOR_, HOR_, 0, (long)N_ * HOR_, (long)N_ * HOR_, B_,
         1.f, 0.f, ACT_NONE);
  k_gcomb<<<g1((long)B_ * N_ * 16), 256, 0, stream>>>(ybuf, gc1, gout1, HOR_, 16, ACT_RELU);
  for (int k = 1; k < 4; ++k)
    gemm(stream, mulL + (long)k * NN, gout1, nullptr, ybuf + (long)k * B_ * N_ * 16,
         N_, 16, N_, N_, 16, 16, 0, (long)N_ * 16, (long)N_ * 16, B_,
         1.f, 0.f, ACT_NONE);
  k_gcomb<<<g1((long)B_ * N_ * HOR_), 256, 0, stream>>>(ybuf, gc2, gout2, 16, HOR_, ACT_SIG);
  k_outT<<<g1((long)B_ * N_ * HOR_), 256, 0, stream>>>(Ff, (float*)d_out);
  k_outT<<<g1((long)B_ * N_ * HOR_), 256, 0, stream>>>(gout2, (float*)d_out + (long)B_ * N_ * HOR_);
}